// MarginDevianceLoss_7533372637492
// MI455X (gfx1250) — compile-verified
//
#include <hip/hip_runtime.h>
#include <math.h>

typedef float v8f __attribute__((ext_vector_type(8)));
typedef float v2f __attribute__((ext_vector_type(2)));

#define DIM 256
#define LDSTR 260   // padded LDS row stride (floats): 4-word skew -> conflict-free frag reads

__device__ __forceinline__ float softplusf(float z) {
    return fmaxf(z, 0.0f) + log1pf(expf(-fabsf(z)));
}

// ---------------------------------------------------------------------------
// Pass 1: per-row statistics of sim = X X^T (pos/neg sums, sumsq, counts, pos min)
// One block per 16-row strip; 8 waves split the 16-column tiles.
// Element classification is branchless (cndmask selects) so the epilogue is
// pure VALU and can co-execute with the WMMA pipe.
// ---------------------------------------------------------------------------
__global__ __launch_bounds__(256) void mdl_stats_kernel(
    const float* __restrict__ X, const int* __restrict__ tg, int n,
    float* __restrict__ posSum, float* __restrict__ posSq, float* __restrict__ posCnt,
    float* __restrict__ negSum, float* __restrict__ negSq, float* __restrict__ negCnt,
    float* __restrict__ posMin)
{
    __shared__ float sA[16 * LDSTR];
    __shared__ float sRed[7][8][16];

    const int tid  = threadIdx.x;
    const int wave = tid >> 5;
    const int lane = tid & 31;
    const int lr   = lane & 15;
    const int hi   = lane >> 4;
    const int i0   = blockIdx.x * 16;

    // Stage 16 x 256 A-strip into LDS (padded rows), float4 per thread x4.
    #pragma unroll
    for (int u = 0; u < 4; ++u) {
        int e   = tid + 256 * u;     // 1024 float4's total
        int row = e >> 6;            // 64 float4 per row
        int c4  = e & 63;
        float4 v = ((const float4*)(X + (size_t)(i0 + row) * DIM))[c4];
        *(float4*)&sA[row * LDSTR + c4 * 4] = v;
    }
    __syncthreads();

    int tRow[8];
    #pragma unroll
    for (int r = 0; r < 8; ++r) tRow[r] = tg[i0 + r + 8 * hi];

    float psum[8], psq[8], pcnt[8], nsum[8], nsq[8], ncnt[8], pmin[8];
    #pragma unroll
    for (int r = 0; r < 8; ++r) {
        psum[r] = psq[r] = pcnt[r] = nsum[r] = nsq[r] = ncnt[r] = 0.0f;
        pmin[r] = __builtin_inff();
    }

    const float* arow = &sA[lr * LDSTR + 2 * hi];
    const int nTiles = n >> 4;

    for (int jt = wave; jt < nTiles; jt += 8) {
        const int j    = jt * 16 + lr;
        const int tCol = tg[j];
        const float* brow = X + (size_t)j * DIM + 2 * hi;

        v8f acc = {};
        #pragma unroll 8
        for (int kc = 0; kc < DIM; kc += 4) {
            v2f a = *(const v2f*)(arow + kc);
            v2f b = *(const v2f*)(brow + kc);
            acc = __builtin_amdgcn_wmma_f32_16x16x4_f32(
                false, a, false, b, (short)0, acc, false, false);
        }

        #pragma unroll
        for (int r = 0; r < 8; ++r) {
            const int   i    = i0 + r + 8 * hi;
            const float s    = acc[r];
            const bool  same = (tRow[r] == tCol);
            const bool  pos  = same && (i != j);
            const float posM = pos  ? 1.0f : 0.0f;
            const float negM = same ? 0.0f : 1.0f;
            psum[r] = fmaf(posM, s, psum[r]);
            psq[r]  = fmaf(posM * s, s, psq[r]);
            pcnt[r] += posM;
            nsum[r] = fmaf(negM, s, nsum[r]);
            nsq[r]  = fmaf(negM * s, s, nsq[r]);
            ncnt[r] += negM;
            pmin[r] = fminf(pmin[r], pos ? s : __builtin_inff());
        }
    }

    // Reduce across the 16 lanes holding one row (xor<16 stays in each half).
    #pragma unroll
    for (int r = 0; r < 8; ++r) {
        #pragma unroll
        for (int off = 8; off; off >>= 1) {
            psum[r] += __shfl_xor(psum[r], off);
            psq[r]  += __shfl_xor(psq[r],  off);
            pcnt[r] += __shfl_xor(pcnt[r], off);
            nsum[r] += __shfl_xor(nsum[r], off);
            nsq[r]  += __shfl_xor(nsq[r],  off);
            ncnt[r] += __shfl_xor(ncnt[r], off);
            pmin[r]  = fminf(pmin[r], __shfl_xor(pmin[r], off));
        }
    }
    if (lr == 0) {
        #pragma unroll
        for (int r = 0; r < 8; ++r) {
            int rl = r + 8 * hi;
            sRed[0][wave][rl] = psum[r];
            sRed[1][wave][rl] = psq[r];
            sRed[2][wave][rl] = pcnt[r];
            sRed[3][wave][rl] = nsum[r];
            sRed[4][wave][rl] = nsq[r];
            sRed[5][wave][rl] = ncnt[r];
            sRed[6][wave][rl] = pmin[r];
        }
    }
    __syncthreads();

    if (tid < 16) {
        float a0 = 0, a1 = 0, a2 = 0, a3 = 0, a4 = 0, a5 = 0, a6 = __builtin_inff();
        #pragma unroll
        for (int w = 0; w < 8; ++w) {
            a0 += sRed[0][w][tid]; a1 += sRed[1][w][tid]; a2 += sRed[2][w][tid];
            a3 += sRed[3][w][tid]; a4 += sRed[4][w][tid]; a5 += sRed[5][w][tid];
            a6  = fminf(a6, sRed[6][w][tid]);
        }
        const int i = i0 + tid;
        posSum[i] = a0; posSq[i] = a1; posCnt[i] = a2;
        negSum[i] = a3; negSq[i] = a4; negCnt[i] = a5;
        posMin[i] = a6;
    }
}

// ---------------------------------------------------------------------------
// Per-row: means, population stds, inter threshold, hard-negative threshold.
// ---------------------------------------------------------------------------
__global__ void mdl_rowstats_kernel(
    const float* __restrict__ posSum, const float* __restrict__ posSq, const float* __restrict__ posCnt,
    const float* __restrict__ negSum, const float* __restrict__ negSq, const float* __restrict__ negCnt,
    const float* __restrict__ posMin,
    float* __restrict__ inter, float* __restrict__ thresh, int n)
{
    int i = blockIdx.x * blockDim.x + threadIdx.x;
    if (i >= n) return;
    float p  = fmaxf(posCnt[i], 1.0f);
    float m  = fmaxf(negCnt[i], 1.0f);
    float pm = posSum[i] / p;
    float nm = negSum[i] / m;
    float ps = sqrtf(fmaxf(posSq[i] / p - pm * pm, 0.0f));
    float ns = sqrtf(fmaxf(negSq[i] / m - nm * nm, 0.0f));
    inter[i]  = 0.8f * (ns * pm + ps * nm) / (ps + ns) + 0.1f;
    thresh[i] = posMin[i] - 0.05f;
}

// ---------------------------------------------------------------------------
// Pass 2: recompute sim tiles, accumulate softplus losses + hard-negative count.
// One softplus per element on a mask-selected argument (pos and neg are
// mutually exclusive), all accumulation branchless.
// ---------------------------------------------------------------------------
__global__ __launch_bounds__(256) void mdl_loss_kernel(
    const float* __restrict__ X, const int* __restrict__ tg, int n,
    const float* __restrict__ inter, const float* __restrict__ thresh,
    const float* __restrict__ posCnt,
    float* __restrict__ rowLoss, float* __restrict__ rowValid)
{
    __shared__ float sA[16 * LDSTR];
    __shared__ float sRed[3][8][16];

    const int tid  = threadIdx.x;
    const int wave = tid >> 5;
    const int lane = tid & 31;
    const int lr   = lane & 15;
    const int hi   = lane >> 4;
    const int i0   = blockIdx.x * 16;

    #pragma unroll
    for (int u = 0; u < 4; ++u) {
        int e   = tid + 256 * u;
        int row = e >> 6;
        int c4  = e & 63;
        float4 v = ((const float4*)(X + (size_t)(i0 + row) * DIM))[c4];
        *(float4*)&sA[row * LDSTR + c4 * 4] = v;
    }
    __syncthreads();

    int   tRow[8];
    float interR[8], threshR[8];
    #pragma unroll
    for (int r = 0; r < 8; ++r) {
        int i = i0 + r + 8 * hi;
        tRow[r]   = tg[i];
        interR[r] = inter[i];
        threshR[r]= thresh[i];
    }

    float pl[8], nl[8], cn[8];
    #pragma unroll
    for (int r = 0; r < 8; ++r) pl[r] = nl[r] = cn[r] = 0.0f;

    const float* arow = &sA[lr * LDSTR + 2 * hi];
    const int nTiles = n >> 4;

    for (int jt = wave; jt < nTiles; jt += 8) {
        const int j    = jt * 16 + lr;
        const int tCol = tg[j];
        const float* brow = X + (size_t)j * DIM + 2 * hi;

        v8f acc = {};
        #pragma unroll 8
        for (int kc = 0; kc < DIM; kc += 4) {
            v2f a = *(const v2f*)(arow + kc);
            v2f b = *(const v2f*)(brow + kc);
            acc = __builtin_amdgcn_wmma_f32_16x16x4_f32(
                false, a, false, b, (short)0, acc, false, false);
        }

        #pragma unroll
        for (int r = 0; r < 8; ++r) {
            const int   i    = i0 + r + 8 * hi;
            const float s    = acc[r];
            const float d    = s - interR[r];
            const bool  same = (tRow[r] == tCol);
            const bool  pos  = same && (i != j);
            const bool  filt = (!same) && (s > threshR[r]);
            // pos and filt are mutually exclusive -> one softplus evaluation.
            const float arg  = same ? (-10.0f * d) : (40.0f * d);
            const float spv  = softplusf(arg);
            pl[r] += pos  ? spv  : 0.0f;
            nl[r] += filt ? spv  : 0.0f;
            cn[r] += filt ? 1.0f : 0.0f;
        }
    }

    #pragma unroll
    for (int r = 0; r < 8; ++r) {
        #pragma unroll
        for (int off = 8; off; off >>= 1) {
            pl[r] += __shfl_xor(pl[r], off);
            nl[r] += __shfl_xor(nl[r], off);
            cn[r] += __shfl_xor(cn[r], off);
        }
    }
    if (lr == 0) {
        #pragma unroll
        for (int r = 0; r < 8; ++r) {
            int rl = r + 8 * hi;
            sRed[0][wave][rl] = pl[r];
            sRed[1][wave][rl] = nl[r];
            sRed[2][wave][rl] = cn[r];
        }
    }
    __syncthreads();

    if (tid < 16) {
        float PL = 0, NL = 0, CN = 0;
        #pragma unroll
        for (int w = 0; w < 8; ++w) {
            PL += sRed[0][w][tid]; NL += sRed[1][w][tid]; CN += sRed[2][w][tid];
        }
        const int i = i0 + tid;
        float p    = fmaxf(posCnt[i], 1.0f);
        float posL = 0.2f  * PL / p;
        float negL = 0.05f * NL / fmaxf(CN, 1.0f);
        bool  valid = (CN > 0.0f);
        rowLoss[i]  = valid ? (posL + negL) : 0.0f;
        rowValid[i] = valid ? 1.0f : 0.0f;
    }
}

// ---------------------------------------------------------------------------
// Final reduction to 4 scalars: loss, prec, pos_d, neg_d.
// ---------------------------------------------------------------------------
__global__ __launch_bounds__(256) void mdl_final_kernel(
    const float* __restrict__ rowLoss, const float* __restrict__ rowValid,
    const float* __restrict__ posSum, const float* __restrict__ posCnt,
    const float* __restrict__ negSum, const float* __restrict__ negCnt,
    float* __restrict__ out, int n)
{
    __shared__ float red[6][256];
    const int tid = threadIdx.x;
    float a0 = 0, a1 = 0, a2 = 0, a3 = 0, a4 = 0, a5 = 0;
    for (int i = tid; i < n; i += 256) {
        a0 += rowLoss[i];  a1 += rowValid[i];
        a2 += posSum[i];   a3 += posCnt[i];
        a4 += negSum[i];   a5 += negCnt[i];
    }
    red[0][tid] = a0; red[1][tid] = a1; red[2][tid] = a2;
    red[3][tid] = a3; red[4][tid] = a4; red[5][tid] = a5;
    __syncthreads();
    for (int off = 128; off; off >>= 1) {
        if (tid < off) {
            #pragma unroll
            for (int s = 0; s < 6; ++s) red[s][tid] += red[s][tid + off];
        }
        __syncthreads();
    }
    if (tid == 0) {
        float fn = (float)n;
        out[0] = red[0][0] / fn;                 // loss
        out[1] = (fn - red[1][0]) / fn;          // prec = mean(~valid)
        out[2] = red[2][0] / fmaxf(red[3][0], 1.0f); // pos_d
        out[3] = red[4][0] / fmaxf(red[5][0], 1.0f); // neg_d
    }
}

extern "C" void kernel_launch(void* const* d_in, const int* in_sizes, int n_in,
                              void* d_out, int out_size, void* d_ws, size_t ws_size,
                              hipStream_t stream) {
    const float* X  = (const float*)d_in[0];
    const int*   tg = (const int*)d_in[1];
    float* out = (float*)d_out;
    float* ws  = (float*)d_ws;

    const int n = in_sizes[1];           // 4096 rows; D fixed at 256

    float* posSum  = ws + 0 * (size_t)n;
    float* posSq   = ws + 1 * (size_t)n;
    float* posCnt  = ws + 2 * (size_t)n;
    float* negSum  = ws + 3 * (size_t)n;
    float* negSq   = ws + 4 * (size_t)n;
    float* negCnt  = ws + 5 * (size_t)n;
    float* posMin  = ws + 6 * (size_t)n;
    float* inter   = ws + 7 * (size_t)n;
    float* thresh  = ws + 8 * (size_t)n;
    float* rowLoss = ws + 9 * (size_t)n;
    float* rowVal  = ws + 10 * (size_t)n;

    mdl_stats_kernel<<<n / 16, 256, 0, stream>>>(
        X, tg, n, posSum, posSq, posCnt, negSum, negSq, negCnt, posMin);
    mdl_rowstats_kernel<<<(n + 255) / 256, 256, 0, stream>>>(
        posSum, posSq, posCnt, negSum, negSq, negCnt, posMin, inter, thresh, n);
    mdl_loss_kernel<<<n / 16, 256, 0, stream>>>(
        X, tg, n, inter, thresh, posCnt, rowLoss, rowVal);
    mdl_final_kernel<<<1, 256, 0, stream>>>(
        rowLoss, rowVal, posSum, posCnt, negSum, negCnt, out, n);
}